// STFTProcessor_46883863003253
// MI455X (gfx1250) — compile-verified
//
#include <hip/hip_runtime.h>
#include <hip/hip_bf16.h>
#include <math.h>

// ---------------- problem constants ----------------
#define Bv    8
#define Tv    8384
#define Ev    256
#define WINv  256
#define STRv  64
#define Fv    129      // WIN/2+1
#define FPAD  144      // F padded to multiple of 16
#define Tp    128      // (T-WIN)/STR+1
#define EM    64       // E/M
#define BM    32       // B*M

// flat strides of X in (B,Tp,F,E) view
#define XS_B  4227072ULL   // Tp*F*E
#define XS_T  33024ULL     // F*E
// same buffer viewed as (BM,Tp,F,EM)
#define RS_BM 1056768ULL   // Tp*F*EM
#define RS_T  8256ULL      // F*EM

typedef float v2f __attribute__((ext_vector_type(2)));
typedef float v8f __attribute__((ext_vector_type(8)));
typedef int   v4i __attribute__((ext_vector_type(4)));

// D(16x16 f32) = A(16x4 f32) x B(4x16 f32) + C  -- full-precision CDNA5 WMMA
#define WMMA_F32(A, B, C) \
  __builtin_amdgcn_wmma_f32_16x16x4_f32(false, (A), false, (B), (short)0, (C), false, false)

// CDNA5 async global->LDS DMA (ASYNCcnt path); fallback to plain loads if absent
#if defined(__has_builtin)
#if __has_builtin(__builtin_amdgcn_global_load_async_to_lds_b128) && \
    __has_builtin(__builtin_amdgcn_s_wait_asynccnt)
#define HAVE_ASYNC_LDS 1
#endif
#endif
// builtin signature: (v4i addrspace(1)*, v4i addrspace(3)*, imm offset, imm cpol)
#define AS_GLB_V4I(p) ((__attribute__((address_space(1))) v4i*)(uintptr_t)(p))
#define AS_LDS_V4I(p) ((__attribute__((address_space(3))) v4i*)(unsigned int)(uintptr_t)(p))

#define PI2F 6.2831853071795864769f

// ---------------- K0: DFT / iDFT constant matrices (window folded in) ----------------
__global__ __launch_bounds__(256) void k0_const(float* __restrict__ CRe, float* __restrict__ CIm,
                                                float* __restrict__ IRe, float* __restrict__ IIm) {
  int i = blockIdx.x * 256 + threadIdx.x;
  if (i >= FPAD * WINv) return;
  {
    // forward: C[f][w], row-major [FPAD][WIN]
    int f = i / WINv, w = i % WINv;
    float cr = 0.f, ci = 0.f;
    if (f < Fv) {
      float win = 0.5f * (1.0f - cosf(PI2F * (float)w / 256.0f));
      float th  = PI2F * (float)((f * w) & 255) / 256.0f;   // exact range reduction
      cr =  win * cosf(th);
      ci = -win * sinf(th);
    }
    CRe[i] = cr; CIm[i] = ci;
  }
  {
    // inverse: I[w][f], row-major [WIN][FPAD]
    int w = i / FPAD, f = i % FPAD;
    float ir = 0.f, ii = 0.f;
    if (f < Fv) {
      float th = PI2F * (float)((f * w) & 255) / 256.0f;
      bool edge = (f == 0) || (f == 128);
      float cf = edge ? 1.0f : 2.0f;
      ir = cf * cosf(th) * (1.0f / 256.0f);
      ii = edge ? 0.0f : (-2.0f * sinf(th) * (1.0f / 256.0f)); // imag of DC/Nyquist ignored (C2R)
    }
    IRe[i] = ir; IIm[i] = ii;
  }
}

// ---------------- K1: STFT as GEMM  X[f,e] = C[f,w] @ frame[w,e] per (b,t) ----------------
// grid(tileGroup=18, bt=1024): same-(b,t) blocks adjacent -> frame stays L2-resident
__global__ __launch_bounds__(256) void k1_stft(const float* __restrict__ x,
                                               const float* __restrict__ CRe,
                                               const float* __restrict__ CIm,
                                               float* __restrict__ Xre, float* __restrict__ Xim) {
  int bt = blockIdx.y;                 // 0..B*Tp-1
  int b = bt >> 7, t = bt & 127;
  int wave = threadIdx.x >> 5, lane = threadIdx.x & 31;
  int half = lane >> 4, l16 = lane & 15;
  int tile = blockIdx.x * 8 + wave;    // 0..143 : 9 f-tiles x 16 e-tiles
  int ft = tile >> 4, et = tile & 15;

  const float* xb = x + ((size_t)b * Tv + (size_t)t * STRv) * Ev + et * 16 + l16;
  const float* arp = CRe + (ft * 16 + l16) * WINv + 2 * half;
  const float* aip = CIm + (ft * 16 + l16) * WINv + 2 * half;

  __builtin_prefetch(xb + (size_t)128 * Ev, 0, 1);   // global_prefetch_b8: back half of frame

  v8f accR = {}; v8f accI = {};
  for (int k0 = 0; k0 < WINv; k0 += 4) {
    v2f are, aim, bfr;
    are.x = arp[k0]; are.y = arp[k0 + 1];
    aim.x = aip[k0]; aim.y = aip[k0 + 1];
    int kr = k0 + 2 * half;
    bfr.x = xb[(size_t)kr * Ev];
    bfr.y = xb[(size_t)(kr + 1) * Ev];
    accR = WMMA_F32(are, bfr, accR);
    accI = WMMA_F32(aim, bfr, accI);
  }
  size_t base = (size_t)b * XS_B + (size_t)t * XS_T;
  int e = et * 16 + l16;
#pragma unroll
  for (int v = 0; v < 8; ++v) {
    int f = ft * 16 + v + 8 * half;
    if (f < Fv) {
      Xre[base + (size_t)f * Ev + e] = accR[v];
      Xim[base + (size_t)f * Ev + e] = accI[v];
    }
  }
}

// ---------------- K2a: compressed[t,e'] = mean over (bm,f) of Re X ----------------
__global__ __launch_bounds__(256) void k2a_compress(const float* __restrict__ Xre,
                                                    float* __restrict__ comp) {
  int t = blockIdx.x;
  int e = threadIdx.x & 63, g = threadIdx.x >> 6;
  __shared__ float red[256];
  float s = 0.f;
  for (int bm = g; bm < BM; bm += 4) {
    size_t base = (size_t)bm * RS_BM + (size_t)t * RS_T + e;
    for (int f = 0; f < Fv; ++f) s += Xre[base + (size_t)f * EM];
  }
  red[threadIdx.x] = s;
  __syncthreads();
  if (g == 0)
    comp[t * EM + e] = (red[e] + red[64 + e] + red[128 + e] + red[192 + e]) * (1.0f / 4128.0f);
}

// ---------------- K2b: h = gelu(compressed @ W1 + b1), exact gelu ----------------
__global__ __launch_bounds__(128) void k2b_mlp1(const float* __restrict__ comp,
                                                const float* __restrict__ W1,
                                                const float* __restrict__ b1,
                                                float* __restrict__ h) {
  int t = blockIdx.x, j = threadIdx.x;
  float acc = b1[j];
  for (int e = 0; e < EM; ++e) acc += comp[t * EM + e] * W1[e * 128 + j];
  h[t * 128 + j] = 0.5f * acc * (1.0f + erff(acc * 0.70710678118654752f));
}

// ---------------- K2c: U = (h @ W2 + b2), 16-t register blocking (W2 read 8x, not 128x) ----
__global__ __launch_bounds__(256) void k2c_mlp2(const float* __restrict__ h,
                                                const float* __restrict__ W2,
                                                const float* __restrict__ b2,
                                                float* __restrict__ Ure, float* __restrict__ Uim) {
  __shared__ float hs[16 * 128];                 // 8 KB
  int tg = blockIdx.y;                           // 8 groups of 16 t
  int c  = blockIdx.x * 256 + threadIdx.x;       // 0..4095
  for (int i = threadIdx.x; i < 16 * 128; i += 256)
    hs[i] = h[(size_t)(tg * 16) * 128 + i];
  __syncthreads();

  float accR[16], accI[16];
  float bre = b2[c], bie = b2[4096 + c];
#pragma unroll
  for (int tt = 0; tt < 16; ++tt) { accR[tt] = bre; accI[tt] = bie; }
  for (int k = 0; k < 128; ++k) {
    float wr = W2[(size_t)k * 8192 + c];
    float wi = W2[(size_t)k * 8192 + 4096 + c];
#pragma unroll
    for (int tt = 0; tt < 16; ++tt) {
      float hv = hs[tt * 128 + k];
      accR[tt] += hv * wr;
      accI[tt] += hv * wi;
    }
  }
#pragma unroll
  for (int tt = 0; tt < 16; ++tt) {
    int t = tg * 16 + tt;
    Ure[(size_t)t * 4096 + c] = accR[tt];
    Uim[(size_t)t * 4096 + c] = accI[tt];
  }
}

// ---------------- K2d: D[f] = sum_n 1/(1 - A_n e^{-i 2 pi f/256}) ----------------
__global__ __launch_bounds__(256) void k2d_dfilter(const float* __restrict__ Ar,
                                                   const float* __restrict__ Ai,
                                                   float* __restrict__ Dre, float* __restrict__ Dim) {
  int f = threadIdx.x;
  if (f >= Fv) return;
  float th = PI2F * (float)f / 256.0f;
  float c = cosf(th), s = sinf(th);
  float sr = 0.f, si = 0.f;
  for (int n = 0; n < 64; ++n) {
    float ar = Ar[n], ai = Ai[n];
    float dr = 1.0f - (ar * c + ai * s);
    float di = ar * s - ai * c;
    float inv = 1.0f / (dr * dr + di * di);
    sr += dr * inv;
    si -= di * inv;
  }
  Dre[f] = sr; Dim[f] = si;
}

// ---------------- K3: complex GEMM  Y = relu((U[t] @ X) * D + bias), in place ----------------
// grid(colGroup=33, t=128)
__global__ __launch_bounds__(256) void k3_einsum(float* __restrict__ Xre, float* __restrict__ Xim,
                                                 const float* __restrict__ Ure,
                                                 const float* __restrict__ Uim,
                                                 const float* __restrict__ Dre,
                                                 const float* __restrict__ Dim,
                                                 const float* __restrict__ br,
                                                 const float* __restrict__ bi) {
  __shared__ float sUre[4096], sUim[4096];       // 32 KB of 320 KB WGP LDS
  int t = blockIdx.y;
#if defined(HAVE_ASYNC_LDS)
  // Async DMA global->LDS: 16B per lane per issue, tracked by ASYNCcnt
  for (int i = threadIdx.x * 4; i < 4096; i += 1024) {
    __builtin_amdgcn_global_load_async_to_lds_b128(
        AS_GLB_V4I(Ure + (size_t)t * 4096 + i), AS_LDS_V4I(&sUre[i]), 0, 0);
    __builtin_amdgcn_global_load_async_to_lds_b128(
        AS_GLB_V4I(Uim + (size_t)t * 4096 + i), AS_LDS_V4I(&sUim[i]), 0, 0);
  }
  __builtin_amdgcn_s_wait_asynccnt(0);
  __syncthreads();
#else
  for (int i = threadIdx.x; i < 4096; i += 256) {
    sUre[i] = Ure[(size_t)t * 4096 + i];
    sUim[i] = Uim[(size_t)t * 4096 + i];
  }
  __syncthreads();
#endif

  int wave = threadIdx.x >> 5, lane = threadIdx.x & 31;
  int half = lane >> 4, l16 = lane & 15;
  int ct = blockIdx.x * 8 + wave;                // column tile over 4128 = 258 tiles
  if (ct >= 258) return;                          // wave-uniform: EXEC stays all-ones

  int j  = ct * 16 + l16;                         // j = bm*129 + f'
  int bm = j / Fv, fp = j % Fv;
  size_t colbase = (size_t)bm * RS_BM + (size_t)t * RS_T + (size_t)fp * EM;

  v8f accRe[4] = {{}, {}, {}, {}};
  v8f accIm[4] = {{}, {}, {}, {}};
  for (int k0 = 0; k0 < EM; k0 += 4) {
    int kr = k0 + 2 * half;
    v2f bre, bim, nbim;
    bre.x = Xre[colbase + kr];     bre.y = Xre[colbase + kr + 1];
    bim.x = Xim[colbase + kr];     bim.y = Xim[colbase + kr + 1];
    nbim.x = -bim.x;               nbim.y = -bim.y;   // f32 WMMA has no A/B negate
#pragma unroll
    for (int rt = 0; rt < 4; ++rt) {
      int arow = rt * 16 + l16;
      v2f are, aim;
      are.x = sUre[arow * EM + kr]; are.y = sUre[arow * EM + kr + 1];
      aim.x = sUim[arow * EM + kr]; aim.y = sUim[arow * EM + kr + 1];
      accRe[rt] = WMMA_F32(are, bre,  accRe[rt]);   // +Ure*Xre
      accRe[rt] = WMMA_F32(aim, nbim, accRe[rt]);   // -Uim*Xim
      accIm[rt] = WMMA_F32(are, bim,  accIm[rt]);   // +Ure*Xim
      accIm[rt] = WMMA_F32(aim, bre,  accIm[rt]);   // +Uim*Xre
    }
  }
  float dr = Dre[fp], di = Dim[fp];
#pragma unroll
  for (int rt = 0; rt < 4; ++rt)
#pragma unroll
    for (int v = 0; v < 8; ++v) {
      int a = rt * 16 + v + 8 * half;
      float sr = accRe[rt][v], si = accIm[rt][v];
      float yr = sr * dr - si * di + br[a];
      float yi = sr * di + si * dr + bi[a];
      Xre[colbase + a] = fmaxf(yr, 0.f);            // own columns only: race-free in place
      Xim[colbase + a] = fmaxf(yi, 0.f);
    }
}

// ---------------- K4: irfft as GEMM + overlap-add (atomicAdd) ----------------
// grid(tileGroup=32, bt=1024): same-(b,t) blocks adjacent -> Y slice stays L2-resident
__global__ __launch_bounds__(256) void k4_irfft(const float* __restrict__ Xre,
                                                const float* __restrict__ Xim,
                                                const float* __restrict__ IRe,
                                                const float* __restrict__ IIm,
                                                float* __restrict__ out) {
  int bt = blockIdx.y;
  int b = bt >> 7, t = bt & 127;
  int wave = threadIdx.x >> 5, lane = threadIdx.x & 31;
  int half = lane >> 4, l16 = lane & 15;
  int tile = blockIdx.x * 8 + wave;               // 0..255 : 16 w-tiles x 16 e-tiles
  int wt = tile >> 4, et = tile & 15;

  size_t base = (size_t)b * XS_B + (size_t)t * XS_T + et * 16 + l16;
  const float* arp = IRe + (wt * 16 + l16) * FPAD + 2 * half;
  const float* aip = IIm + (wt * 16 + l16) * FPAD + 2 * half;

  v8f acc = {};
  for (int k0 = 0; k0 < FPAD; k0 += 4) {          // rows f=129..143: A-cols are zero
    int kr = k0 + 2 * half;
    v2f are, aim, bre, bim;
    are.x = arp[k0]; are.y = arp[k0 + 1];
    aim.x = aip[k0]; aim.y = aip[k0 + 1];
    bre.x = Xre[base + (size_t)kr * Ev]; bre.y = Xre[base + (size_t)(kr + 1) * Ev];
    bim.x = Xim[base + (size_t)kr * Ev]; bim.y = Xim[base + (size_t)(kr + 1) * Ev];
    acc = WMMA_F32(are, bre, acc);
    acc = WMMA_F32(aim, bim, acc);
  }
  int e = et * 16 + l16;
  size_t ob = ((size_t)b * Tv + (size_t)t * STRv) * Ev + e;
#pragma unroll
  for (int v = 0; v < 8; ++v) {
    int w = wt * 16 + v + 8 * half;
    atomicAdd(&out[ob + (size_t)w * Ev], acc[v]);
  }
}

// ---------------- launch ----------------
extern "C" void kernel_launch(void* const* d_in, const int* in_sizes, int n_in,
                              void* d_out, int out_size, void* d_ws, size_t ws_size,
                              hipStream_t stream) {
  (void)in_sizes; (void)n_in; (void)ws_size;
  const float* x  = (const float*)d_in[0];
  const float* W1 = (const float*)d_in[1];
  const float* b1 = (const float*)d_in[2];
  const float* W2 = (const float*)d_in[3];
  const float* b2 = (const float*)d_in[4];
  const float* Ar = (const float*)d_in[5];
  const float* Ai = (const float*)d_in[6];
  const float* br = (const float*)d_in[7];
  const float* bi = (const float*)d_in[8];
  float* out = (float*)d_out;
  float* ws  = (float*)d_ws;
  (void)out;

  const size_t XN = (size_t)Bv * XS_B;            // 33,816,576 floats
  float* Xre  = ws;                 size_t off = XN;
  float* Xim  = ws + off;           off += XN;
  float* CRe  = ws + off;           off += FPAD * WINv;
  float* CIm  = ws + off;           off += FPAD * WINv;
  float* IRe  = ws + off;           off += WINv * FPAD;
  float* IIm  = ws + off;           off += WINv * FPAD;
  float* comp = ws + off;           off += Tp * EM;
  float* h    = ws + off;           off += Tp * 128;
  float* Ure  = ws + off;           off += (size_t)Tp * 4096;
  float* Uim  = ws + off;           off += (size_t)Tp * 4096;
  float* Dre  = ws + off;           off += 256;
  float* Dim  = ws + off;           off += 256;
  // total ~68.9M floats (~276 MB); region after Xim absorbs the benign
  // zero-weighted over-reads past Xim in K4.

  (void)hipMemsetAsync(d_out, 0, (size_t)out_size * sizeof(float), stream);

  k0_const    <<<FPAD, 256, 0, stream>>>(CRe, CIm, IRe, IIm);
  k2d_dfilter <<<1, 256, 0, stream>>>(Ar, Ai, Dre, Dim);
  k1_stft     <<<dim3(18, Bv * Tp), 256, 0, stream>>>(x, CRe, CIm, Xre, Xim);
  k2a_compress<<<Tp, 256, 0, stream>>>(Xre, comp);
  k2b_mlp1    <<<Tp, 128, 0, stream>>>(comp, W1, b1, h);
  k2c_mlp2    <<<dim3(16, 8), 256, 0, stream>>>(h, W2, b2, Ure, Uim);
  k3_einsum   <<<dim3(33, Tp), 256, 0, stream>>>(Xre, Xim, Ure, Uim, Dre, Dim, br, bi);
  k4_irfft    <<<dim3(32, Bv * Tp), 256, 0, stream>>>(Xre, Xim, IRe, IIm, out);
}